// QuantLinear_65764539236729
// MI455X (gfx1250) — compile-verified
//
#include <hip/hip_runtime.h>

// CDNA5 / gfx1250, wave32. Uses V_WMMA_F32_16X16X4_F32 (the K=4 f32 WMMA)
// to implement the per-chunk 4-wide dot products of QuantLinear.

typedef __attribute__((ext_vector_type(2))) float v2f;
typedef __attribute__((ext_vector_type(8))) float v8f;

#define WAVES_PER_BLOCK 8
#define TILES_PER_WAVE  4   // 16-batch-row tiles per wave
// rows per block = 16 * 8 * 4 = 512

__device__ __forceinline__ float signf3(float x) {
    return (x > 0.0f) ? 1.0f : ((x < 0.0f) ? -1.0f : 0.0f);
}

__global__ __launch_bounds__(256)
void quantlinear_wmma_kernel(const float* __restrict__ Input,   // [B,16]
                             const float* __restrict__ Weight,  // [32,16]
                             const float* __restrict__ pInMin,  // [1]
                             const float* __restrict__ pInMax,  // [1]
                             float* __restrict__ Out,           // [B,32]
                             int Btotal)
{
    const int lane = threadIdx.x & 31;
    const int wave = threadIdx.x >> 5;
    const int n    = lane & 15;   // N column (B/D) or M row (A)
    const int half = lane >> 4;   // selects K-half for A/B, M-half for D

    const float InMin = pInMin[0];
    const float InMax = pInMax[0];
    const float scale = InMax - InMin;
    const float qmul  = 15.0f / scale;
    const float inv15 = 1.0f / 15.0f;

    // ---- loop-invariant B-matrices from sign(weight): wP = relu(s), wM = relu(-s)
    // B layout (4x16 f32): lane(half,n) holds W[K = 2*half + {0,1}][N = n]
    v2f bP[2][4], bM[2][4];
    float bias[2];
    #pragma unroll
    for (int t = 0; t < 2; ++t) {
        const int o = t * 16 + n;           // output column 0..31
        float wsum = 0.0f;
        #pragma unroll
        for (int c = 0; c < 4; ++c) {
            const int col = c * 4 + half * 2;
            const float w0 = Weight[o * 16 + col + 0];
            const float w1 = Weight[o * 16 + col + 1];
            const float s0 = signf3(w0);
            const float s1 = signf3(w1);
            v2f p; p.x = fmaxf(s0, 0.0f);  p.y = fmaxf(s1, 0.0f);
            v2f m; m.x = fmaxf(-s0, 0.0f); m.y = fmaxf(-s1, 0.0f);
            bP[t][c] = p;
            bM[t][c] = m;
            wsum += s0 + s1;
        }
        // partner lane (half^1) holds the complementary K-halves of column o
        wsum += __shfl_xor(wsum, 16);
        bias[t] = InMin * wsum;
    }

    const v8f zeroC = {};

    const int tile0 = (blockIdx.x * WAVES_PER_BLOCK + wave) * TILES_PER_WAVE;

    for (int it = 0; it < TILES_PER_WAVE; ++it) {
        const int rowBase = (tile0 + it) * 16;
        if (rowBase >= Btotal) break;      // uniform per wave: EXEC stays all-1s

        // ---- A-matrices: q = clip(round((x-InMin)/scale*15), 0, 15)
        // A layout (16x4 f32): lane(half,n) holds Q[M = n][K = 2*half + {0,1}]
        const float* rowPtr = Input + (size_t)(rowBase + n) * 16;
        v2f a[4];
        #pragma unroll
        for (int c = 0; c < 4; ++c) {
            const v2f x = __builtin_nontemporal_load(
                (const v2f*)(rowPtr + c * 4 + half * 2));
            v2f q;
            q.x = fminf(fmaxf(rintf((x.x - InMin) * qmul), 0.0f), 15.0f);
            q.y = fminf(fmaxf(rintf((x.y - InMin) * qmul), 0.0f), 15.0f);
            a[c] = q;
        }

        v8f acc0 = zeroC;   // N tile 0 (out cols 0..15)
        v8f acc1 = zeroC;   // N tile 1 (out cols 16..31)

        // ---- 4 chunks x 2 N-tiles x 2 signs of 16x16x4 f32 WMMA,
        //      nonlinearity clip(round(y/15),-4,4) between chunks.
        #pragma unroll
        for (int c = 0; c < 4; ++c) {
            #pragma unroll
            for (int t = 0; t < 2; ++t) {
                v8f yP = __builtin_amdgcn_wmma_f32_16x16x4_f32(
                    false, a[c], false, bP[t][c], (short)0, zeroC, false, false);
                v8f yM = __builtin_amdgcn_wmma_f32_16x16x4_f32(
                    false, a[c], false, bM[t][c], (short)0, zeroC, false, false);
                #pragma unroll
                for (int v = 0; v < 8; ++v) {
                    const float tp = fminf(fmaxf(rintf(yP[v] *  inv15), -4.0f), 4.0f);
                    const float tm = fminf(fmaxf(rintf(yM[v] * -inv15), -4.0f), 4.0f);
                    const float s  = tp + tm;
                    if (t == 0) acc0[v] += s; else acc1[v] += s;
                }
            }
        }

        // ---- D layout: VGPR v, lane(half,n) = element [M = v + 8*half][N = n]
        // store Out[row][t*16 + n]; each half-wave writes 64B contiguous per row
        #pragma unroll
        for (int v = 0; v < 8; ++v) {
            const size_t row = (size_t)(rowBase + v + half * 8);
            __builtin_nontemporal_store(acc0[v] * scale + bias[0],
                                        Out + row * 32 + 0  + n);
            __builtin_nontemporal_store(acc1[v] * scale + bias[1],
                                        Out + row * 32 + 16 + n);
        }
    }
}

extern "C" void kernel_launch(void* const* d_in, const int* in_sizes, int n_in,
                              void* d_out, int out_size, void* d_ws, size_t ws_size,
                              hipStream_t stream) {
    (void)n_in; (void)d_ws; (void)ws_size;
    const float* Input  = (const float*)d_in[0];   // [B,16] fp32
    const float* Weight = (const float*)d_in[1];   // [32,16] fp32
    const float* pInMin = (const float*)d_in[2];   // [1]
    const float* pInMax = (const float*)d_in[3];   // [1]
    float* Out = (float*)d_out;                    // [B,32] fp32

    const int B = in_sizes[0] / 16;                // 524288
    const int rowsPerBlock = 16 * WAVES_PER_BLOCK * TILES_PER_WAVE;  // 512
    const int grid = (B + rowsPerBlock - 1) / rowsPerBlock;          // 1024

    quantlinear_wmma_kernel<<<grid, 256, 0, stream>>>(Input, Weight, pInMin,
                                                      pInMax, Out, B);
}